// HWTV3D_20409684590799
// MI455X (gfx1250) — compile-verified
//
#include <hip/hip_runtime.h>
#include <math.h>

// Problem dims (fixed by reference)
#define NB 4
#define CC 31
#define HH 128
#define WW 128
#define NE (NB*CC*HH*WW)          // 2,031,616 elements
#define SLAB (HH*WW)              // 16384
#define LINES (NB*CC*HH)          // 15872 lines for W-axis transform
#define LDSP 132                  // LDS tile pitch (132 % 64 banks = 4 -> conflict-free)

typedef __attribute__((ext_vector_type(2))) float v2f;
typedef __attribute__((ext_vector_type(8))) float v8f;

// ---------------------------------------------------------------------------
// WMMA f32 16x16x4: D = A(16x4) * B(4x16) + C
// ---------------------------------------------------------------------------
__device__ __forceinline__ v8f wmma4(v2f a, v2f b, v8f c) {
#if defined(__HIP_DEVICE_COMPILE__)
  return __builtin_amdgcn_wmma_f32_16x16x4_f32(false, a, false, b, (short)0, c,
                                               false, false);
#else
  (void)a; (void)b;
  return c;
#endif
}

// Fragment fetch from an LDS tile T[row][n] with pitch LDSP.
// Works for both operands:
//  A (16x4): row = M index (lane&15), K = k0 + 2*(lane>=16) (+1 for .y)
//  B (4x16): row = N index (lane&15), K = same split
__device__ __forceinline__ v2f frag2(const float* __restrict__ T, int k0, int lane) {
  int r = lane & 15;
  int n = k0 + ((lane >> 4) << 1);
  v2f v;
  v.x = T[r * LDSP + n];
  v.y = T[r * LDSP + n + 1];
  return v;
}

// ---------------------------------------------------------------------------
// Table init: DFT-128 fwd (Wr=cos, Wi=-sin), inverse scaled (Vr=cos/128,
// Vi=sin/128); DFT-31 fwd/inv; C-axis Laplacian eigenvalues dc.
// ---------------------------------------------------------------------------
__global__ void k_init_tables(float* Wr, float* Wi, float* Vr, float* Vi,
                              float* Cfr, float* Cfi, float* Cvr, float* Cvi,
                              float* dcT) {
  const double PI2 = 6.283185307179586476925286766559;
  int tid = blockIdx.x * blockDim.x + threadIdx.x;
  if (tid < 128 * 128) {
    int k = tid >> 7, n = tid & 127;
    double th = PI2 * (double)((k * n) & 127) / 128.0;
    double ct = cos(th), st = sin(th);
    Wr[tid] = (float)ct;
    Wi[tid] = (float)(-st);
    Vr[tid] = (float)(ct / 128.0);
    Vi[tid] = (float)(st / 128.0);
  }
  if (tid < 31 * 31) {
    int k = tid / 31, n = tid % 31;
    double th = PI2 * (double)((k * n) % 31) / 31.0;
    double ct = cos(th), st = sin(th);
    Cfr[tid] = (float)ct;
    Cfi[tid] = (float)(-st);
    Cvr[tid] = (float)(ct / 31.0);
    Cvi[tid] = (float)(st / 31.0);
  }
  if (tid < 31) {
    double th = PI2 * (double)tid / 31.0;
    dcT[tid] = (float)(2.0 - 2.0 * cos(th));
  }
}

// ---------------------------------------------------------------------------
// State init: Z = Y, X = Y, duals = 0
// ---------------------------------------------------------------------------
__global__ void k_init_state(const float* __restrict__ Y, float* __restrict__ Z,
                             float* __restrict__ X, float* __restrict__ G1,
                             float* __restrict__ G21, float* __restrict__ G22,
                             float* __restrict__ G23) {
  int i = blockIdx.x * blockDim.x + threadIdx.x;
  if (i >= NE) return;
  float y = Y[i];
  Z[i] = y; X[i] = y;
  G1[i] = 0.f; G21[i] = 0.f; G22[i] = 0.f; G23[i] = 0.f;
}

// soft-threshold(v - g/mu2) + g/mu2
__device__ __forceinline__ float admm_term(float v, float g, float inv2, float thr) {
  float x = v - g * inv2;
  float s = copysignf(fmaxf(fabsf(x) - thr, 0.f), x);
  return s + g * inv2;
}

// ---------------------------------------------------------------------------
// numer = mu1*X + G1 + mu2*( DhT(th) + DvT(tv) + DsT(ts) )
// ---------------------------------------------------------------------------
__global__ void k_numer(const float* __restrict__ Z, const float* __restrict__ X,
                        const float* __restrict__ G1, const float* __restrict__ G21,
                        const float* __restrict__ G22, const float* __restrict__ G23,
                        float* __restrict__ out, float mu1, float mu2, float lam) {
  int idx = blockIdx.x * blockDim.x + threadIdx.x;
  if (idx >= NE) return;
  int w = idx & 127;
  int h = (idx >> 7) & 127;
  int t = idx >> 14;
  int c = t % CC;

  int iw_p = idx - w + ((w + 1) & 127);
  int iw_m = idx - w + ((w + 127) & 127);
  int ih_p = idx + ((h == 127) ? -127 * 128 : 128);
  int ih_m = idx + ((h == 0) ? 127 * 128 : -128);
  int ic_p = idx + ((c == CC - 1) ? -(CC - 1) * SLAB : SLAB);
  int ic_m = idx + ((c == 0) ? (CC - 1) * SLAB : -SLAB);

  float zc = Z[idx];
  float inv2 = 1.0f / mu2;
  float thr = lam * inv2;

  float th_c = admm_term(zc - Z[iw_p], G21[idx], inv2, thr);
  float tv_c = admm_term(zc - Z[ih_p], G22[idx], inv2, thr);
  float ts_c = admm_term(zc - Z[ic_p], G23[idx], inv2, thr);
  float th_m = admm_term(Z[iw_m] - zc, G21[iw_m], inv2, thr);
  float tv_m = admm_term(Z[ih_m] - zc, G22[ih_m], inv2, thr);
  float ts_m = admm_term(Z[ic_m] - zc, G23[ic_m], inv2, thr);

  out[idx] = mu1 * X[idx] + G1[idx] +
             mu2 * ((th_c - th_m) + (tv_c - tv_m) + (ts_c - ts_m));
}

// ---------------------------------------------------------------------------
// Forward DFT-128 along W, real -> complex.
// Block = 4 waves x 16 lines each (64 lines). All waves share one A-tile per
// k-tile iteration (staged in LDS); the K-loop is ds_load + v_wmma only.
// ---------------------------------------------------------------------------
__global__ void k_fftW_r2c(const float* __restrict__ in,
                           float* __restrict__ outR, float* __restrict__ outI,
                           const float* __restrict__ Wr, const float* __restrict__ Wi) {
  __shared__ float data[4][16 * LDSP];
  __shared__ float at_r[16 * LDSP];
  __shared__ float at_i[16 * LDSP];
  int tid = threadIdx.x, lane = tid & 31, wave = tid >> 5;
  int lblock = blockIdx.x * 64;

  // stage 64 lines x 128 samples (coalesced along n)
  for (int i = tid; i < 64 * 128; i += 128) {
    int line = i >> 7, n = i & 127;
    data[line >> 4][(line & 15) * LDSP + n] = in[(size_t)(lblock + line) * 128 + n];
  }
  const float* myd = data[wave];
  int col = lane & 15, rb = (lane >> 4) << 3;
  size_t lr = (size_t)(lblock + wave * 16 + col) * 128;

  for (int kt = 0; kt < 8; ++kt) {
    __syncthreads();   // data ready / previous compute done
    for (int i = tid; i < 16 * 128; i += 128) {
      int r = i >> 7, n = i & 127;
      int src = (kt * 16 + r) * 128 + n;
      at_r[r * LDSP + n] = Wr[src];
      at_i[r * LDSP + n] = Wi[src];
    }
    if (kt < 7 && tid < 16) {   // warm next tile rows into L2 (global_prefetch)
      __builtin_prefetch(&Wr[(kt + 1) * 16 * 128 + tid * 128], 0, 0);
      __builtin_prefetch(&Wi[(kt + 1) * 16 * 128 + tid * 128], 0, 0);
    }
    __syncthreads();

    v8f aR = {}, aI = {};
#pragma unroll
    for (int k0 = 0; k0 < 128; k0 += 4) {
      v2f b = frag2(myd, k0, lane);
      aR = wmma4(frag2(at_r, k0, lane), b, aR);
      aI = wmma4(frag2(at_i, k0, lane), b, aI);
    }
#pragma unroll
    for (int j = 0; j < 8; ++j) {
      int k = kt * 16 + rb + j;
      outR[lr + k] = aR[j];
      outI[lr + k] = aI[j];
    }
  }
}

// ---------------------------------------------------------------------------
// Complex DFT-128 along H (fwd or inv via table choice).
// Fr = Tr*xr - Ti*xi (two accumulators, subtract at store); Fi = Tr*xi + Ti*xr
// Block = 4 waves x 16 w-lines; blockIdx = slab*2 + half.
// ---------------------------------------------------------------------------
__global__ void k_fftH_c2c(const float* __restrict__ inR, const float* __restrict__ inI,
                           float* __restrict__ outR, float* __restrict__ outI,
                           const float* __restrict__ Tr, const float* __restrict__ Ti) {
  __shared__ float dr[4][16 * LDSP];
  __shared__ float di[4][16 * LDSP];
  __shared__ float at_r[16 * LDSP];
  __shared__ float at_i[16 * LDSP];
  int tid = threadIdx.x, lane = tid & 31, wave = tid >> 5;
  int slab = blockIdx.x >> 1;
  int wbase = (blockIdx.x & 1) * 64;
  size_t base = (size_t)slab * SLAB;

  // stage 64 w-lines x 128 h-samples (coalesced along w)
  for (int i = tid; i < 64 * 128; i += 128) {
    int wl = i & 63, hh = i >> 6;
    size_t g = base + (size_t)hh * 128 + wbase + wl;
    dr[wl >> 4][(wl & 15) * LDSP + hh] = inR[g];
    di[wl >> 4][(wl & 15) * LDSP + hh] = inI[g];
  }
  const float* mydr = dr[wave];
  const float* mydi = di[wave];
  int col = lane & 15, rb = (lane >> 4) << 3;
  int wout = wbase + wave * 16 + col;

  for (int kt = 0; kt < 8; ++kt) {
    __syncthreads();
    for (int i = tid; i < 16 * 128; i += 128) {
      int r = i >> 7, n = i & 127;
      int src = (kt * 16 + r) * 128 + n;
      at_r[r * LDSP + n] = Tr[src];
      at_i[r * LDSP + n] = Ti[src];
    }
    if (kt < 7 && tid < 16) {
      __builtin_prefetch(&Tr[(kt + 1) * 16 * 128 + tid * 128], 0, 0);
      __builtin_prefetch(&Ti[(kt + 1) * 16 * 128 + tid * 128], 0, 0);
    }
    __syncthreads();

    v8f aP = {}, aM = {}, aJ = {};
#pragma unroll
    for (int k0 = 0; k0 < 128; k0 += 4) {
      v2f br = frag2(mydr, k0, lane);
      v2f bi = frag2(mydi, k0, lane);
      v2f tr = frag2(at_r, k0, lane);
      v2f ti = frag2(at_i, k0, lane);
      aP = wmma4(tr, br, aP);   // Tr*xr
      aM = wmma4(ti, bi, aM);   // Ti*xi
      aJ = wmma4(tr, bi, aJ);   // Tr*xi
      aJ = wmma4(ti, br, aJ);   // + Ti*xr
    }
#pragma unroll
    for (int j = 0; j < 8; ++j) {
      int kh = kt * 16 + rb + j;
      size_t o = base + (size_t)kh * 128 + wout;
      outR[o] = aP[j] - aM[j];
      outI[o] = aJ[j];
    }
  }
}

// ---------------------------------------------------------------------------
// Inverse DFT-128 along W, complex -> real part only.
// Re = Vr*xr - Vi*xi (two accumulators, subtract at store).
// ---------------------------------------------------------------------------
__global__ void k_fftW_c2r(const float* __restrict__ inR, const float* __restrict__ inI,
                           float* __restrict__ outZ,
                           const float* __restrict__ Vr, const float* __restrict__ Vi) {
  __shared__ float dr[4][16 * LDSP];
  __shared__ float di[4][16 * LDSP];
  __shared__ float at_r[16 * LDSP];
  __shared__ float at_i[16 * LDSP];
  int tid = threadIdx.x, lane = tid & 31, wave = tid >> 5;
  int lblock = blockIdx.x * 64;

  for (int i = tid; i < 64 * 128; i += 128) {
    int line = i >> 7, n = i & 127;
    size_t g = (size_t)(lblock + line) * 128 + n;
    dr[line >> 4][(line & 15) * LDSP + n] = inR[g];
    di[line >> 4][(line & 15) * LDSP + n] = inI[g];
  }
  const float* mydr = dr[wave];
  const float* mydi = di[wave];
  int col = lane & 15, rb = (lane >> 4) << 3;
  size_t lr = (size_t)(lblock + wave * 16 + col) * 128;

  for (int kt = 0; kt < 8; ++kt) {
    __syncthreads();
    for (int i = tid; i < 16 * 128; i += 128) {
      int r = i >> 7, n = i & 127;
      int src = (kt * 16 + r) * 128 + n;
      at_r[r * LDSP + n] = Vr[src];
      at_i[r * LDSP + n] = Vi[src];
    }
    if (kt < 7 && tid < 16) {
      __builtin_prefetch(&Vr[(kt + 1) * 16 * 128 + tid * 128], 0, 0);
      __builtin_prefetch(&Vi[(kt + 1) * 16 * 128 + tid * 128], 0, 0);
    }
    __syncthreads();

    v8f aP = {}, aM = {};
#pragma unroll
    for (int k0 = 0; k0 < 128; k0 += 4) {
      aP = wmma4(frag2(at_r, k0, lane), frag2(mydr, k0, lane), aP);
      aM = wmma4(frag2(at_i, k0, lane), frag2(mydi, k0, lane), aM);
    }
#pragma unroll
    for (int j = 0; j < 8; ++j) {
      int n = kt * 16 + rb + j;
      outZ[lr + n] = aP[j] - aM[j];
    }
  }
}

// ---------------------------------------------------------------------------
// Fused: forward DFT-31 along C, divide by (mu1 + mu2*(dc+dh+dw)), inverse
// DFT-31. Whole column in registers; tables cached in LDS.
// ---------------------------------------------------------------------------
__global__ void k_specC(float* __restrict__ R, float* __restrict__ I,
                        float mu1, float mu2,
                        const float* __restrict__ Cfr, const float* __restrict__ Cfi,
                        const float* __restrict__ Cvr, const float* __restrict__ Cvi,
                        const float* __restrict__ dcT) {
  __shared__ float sfr[31 * 31], sfi[31 * 31], svr[31 * 31], svi[31 * 31], sdc[31];
  int tid = threadIdx.x;
  for (int i = tid; i < 31 * 31; i += blockDim.x) {
    sfr[i] = Cfr[i]; sfi[i] = Cfi[i]; svr[i] = Cvr[i]; svi[i] = Cvi[i];
  }
  if (tid < 31) sdc[tid] = dcT[tid];
  __syncthreads();

  int col = blockIdx.x * blockDim.x + tid;
  if (col >= NB * SLAB) return;
  int n = col >> 14;
  int rem = col & (SLAB - 1);
  int kh = rem >> 7, kw = rem & 127;
  size_t base = (size_t)n * CC * SLAB + rem;

  float zr[CC], zi[CC];
#pragma unroll
  for (int c = 0; c < CC; ++c) {
    zr[c] = R[base + (size_t)c * SLAB];
    zi[c] = I[base + (size_t)c * SLAB];
  }

  const float PI2 = 6.28318530717958647692f;
  float dh = 2.f - 2.f * __cosf(PI2 * (float)kh * (1.0f / 128.0f));
  float dw = 2.f - 2.f * __cosf(PI2 * (float)kw * (1.0f / 128.0f));
  float s = mu1 + mu2 * (dh + dw);

  float fr[CC], fi[CC];
#pragma unroll
  for (int k = 0; k < CC; ++k) {
    float ar = 0.f, ai = 0.f;
#pragma unroll
    for (int c = 0; c < CC; ++c) {
      float trr = sfr[k * 31 + c], tii = sfi[k * 31 + c];
      ar += trr * zr[c] - tii * zi[c];
      ai += trr * zi[c] + tii * zr[c];
    }
    float inv = 1.0f / (s + mu2 * sdc[k]);
    fr[k] = ar * inv;
    fi[k] = ai * inv;
  }
#pragma unroll
  for (int c = 0; c < CC; ++c) {
    float ar = 0.f, ai = 0.f;
#pragma unroll
    for (int k = 0; k < CC; ++k) {
      float trr = svr[c * 31 + k], tii = svi[c * 31 + k];
      ar += trr * fr[k] - tii * fi[k];
      ai += trr * fi[k] + tii * fr[k];
    }
    R[base + (size_t)c * SLAB] = ar;
    I[base + (size_t)c * SLAB] = ai;
  }
}

// ---------------------------------------------------------------------------
// X / dual updates (U* recomputed from OLD Z, old duals)
// ---------------------------------------------------------------------------
__global__ void k_update(const float* __restrict__ Zo, const float* __restrict__ Zn,
                         const float* __restrict__ Y, const float* __restrict__ inW,
                         float* __restrict__ X, float* __restrict__ G1,
                         float* __restrict__ G21, float* __restrict__ G22,
                         float* __restrict__ G23,
                         float mu1, float mu2, float lam) {
  int idx = blockIdx.x * blockDim.x + threadIdx.x;
  if (idx >= NE) return;
  int w = idx & 127;
  int h = (idx >> 7) & 127;
  int t = idx >> 14;
  int c = t % CC;
  int iw_p = idx - w + ((w + 1) & 127);
  int ih_p = idx + ((h == 127) ? -127 * 128 : 128);
  int ic_p = idx + ((c == CC - 1) ? -(CC - 1) * SLAB : SLAB);

  float inv2 = 1.0f / mu2;
  float thr = lam * inv2;

  float zo = Zo[idx];
  float g21 = G21[idx], g22 = G22[idx], g23 = G23[idx];
  float uhx = (zo - Zo[iw_p]) - g21 * inv2;
  float uvx = (zo - Zo[ih_p]) - g22 * inv2;
  float usx = (zo - Zo[ic_p]) - g23 * inv2;
  float Uh = copysignf(fmaxf(fabsf(uhx) - thr, 0.f), uhx);
  float Uv = copysignf(fmaxf(fabsf(uvx) - thr, 0.f), uvx);
  float Us = copysignf(fmaxf(fabsf(usx) - thr, 0.f), usx);

  float zn = Zn[idx];
  float dhn = zn - Zn[iw_p];
  float dvn = zn - Zn[ih_p];
  float dsn = zn - Zn[ic_p];

  float g1 = G1[idx];
  float wgt = inW[idx];
  float xn = (wgt * Y[idx] + mu1 * zn - g1) / (wgt + mu1);
  X[idx] = xn;
  G1[idx] = g1 + mu1 * (xn - zn);
  G21[idx] = g21 + mu2 * (Uh - dhn);
  G22[idx] = g22 + mu2 * (Uv - dvn);
  G23[idx] = g23 + mu2 * (Us - dsn);
}

// ---------------------------------------------------------------------------
// Host orchestration
// ---------------------------------------------------------------------------
extern "C" void kernel_launch(void* const* d_in, const int* in_sizes, int n_in,
                              void* d_out, int out_size, void* d_ws, size_t ws_size,
                              hipStream_t stream) {
  (void)in_sizes; (void)n_in; (void)out_size; (void)ws_size;
  const float* Y = (const float*)d_in[0];
  const float* inWp = (const float*)d_in[1];
  float* ws = (float*)d_ws;

  const size_t ne = (size_t)NE;
  float* Za  = ws;
  float* Zb  = Za  + ne;
  float* X   = Zb  + ne;
  float* G1  = X   + ne;
  float* G21 = G1  + ne;
  float* G22 = G21 + ne;
  float* G23 = G22 + ne;
  float* PR  = G23 + ne;
  float* PI_ = PR  + ne;
  float* QR  = PI_ + ne;
  float* QI  = QR  + ne;
  float* tab = QI  + ne;
  float* Wr  = tab;            float* Wi  = Wr  + 16384;
  float* Vr  = Wi  + 16384;    float* Vi  = Vr  + 16384;
  float* Cfr = Vi  + 16384;    float* Cfi = Cfr + 961;
  float* Cvr = Cfi + 961;      float* Cvi = Cvr + 961;
  float* dcT = Cvi + 961;

  k_init_tables<<<64, 256, 0, stream>>>(Wr, Wi, Vr, Vi, Cfr, Cfi, Cvr, Cvi, dcT);
  k_init_state<<<NE / 256, 256, 0, stream>>>(Y, Za, X, G1, G21, G22, G23);

  float mu1 = 0.1f, mu2 = 0.1f;
  const float lam = 0.1f;
  float* Zin = Za;
  float* Zout = Zb;

  for (int it = 0; it < 20; ++it) {
    k_numer<<<NE / 256, 256, 0, stream>>>(Zin, X, G1, G21, G22, G23, PR,
                                          mu1, mu2, lam);
    k_fftW_r2c<<<LINES / 64, 128, 0, stream>>>(PR, QR, QI, Wr, Wi);
    k_fftH_c2c<<<(NB * CC) * 2, 128, 0, stream>>>(QR, QI, PR, PI_, Wr, Wi);
    k_specC<<<(NB * SLAB) / 256, 256, 0, stream>>>(PR, PI_, mu1, mu2,
                                                   Cfr, Cfi, Cvr, Cvi, dcT);
    k_fftH_c2c<<<(NB * CC) * 2, 128, 0, stream>>>(PR, PI_, QR, QI, Vr, Vi);
    k_fftW_c2r<<<LINES / 64, 128, 0, stream>>>(QR, QI, Zout, Vr, Vi);
    k_update<<<NE / 256, 256, 0, stream>>>(Zin, Zout, Y, inWp, X, G1, G21, G22,
                                           G23, mu1, mu2, lam);
    float* tmp = Zin; Zin = Zout; Zout = tmp;
    mu1 *= 1.05f;
    mu2 *= 1.05f;
  }

  hipMemcpyAsync(d_out, Zin, ne * sizeof(float), hipMemcpyDeviceToDevice, stream);
}